// GCNModel_57655640981804
// MI455X (gfx1250) — compile-verified
//
#include <hip/hip_runtime.h>
#include <math.h>

#define NN      100000
#define NE      1600000
#define DHID    128

typedef float v2f __attribute__((ext_vector_type(2)));
typedef float v8f __attribute__((ext_vector_type(8)));

__global__ __launch_bounds__(256) void k_init_deg(float* __restrict__ deg, int n) {
    int i = blockIdx.x * 256 + threadIdx.x;
    if (i < n) deg[i] = 1.0f;               // self-loop contributes 1
}

__global__ __launch_bounds__(256) void k_deg_atomic(const int* __restrict__ dst,
                                                    float* __restrict__ deg, int e) {
    int i = blockIdx.x * 256 + threadIdx.x;
    if (i < e)
        __hip_atomic_fetch_add(&deg[dst[i]], 1.0f,
                               __ATOMIC_RELAXED, __HIP_MEMORY_SCOPE_AGENT);
}

__global__ __launch_bounds__(256) void k_dinv(float* __restrict__ deg, int n) {
    int i = blockIdx.x * 256 + threadIdx.x;
    if (i < n) deg[i] = rsqrtf(deg[i]);     // deg >= 1 always
}

// Out[M x 128] = X[M x 128] @ W[128 x 128] via V_WMMA_F32_16X16X4_F32.
// One wave = one 16x16 tile; block = 8 waves = 16-row strip across 128 cols.
__global__ __launch_bounds__(256) void k_gemm_wmma(const float* __restrict__ X,
                                                   const float* __restrict__ W,
                                                   float* __restrict__ Out) {
    const int wave  = threadIdx.x >> 5;
    const int lane  = threadIdx.x & 31;
    const int l15   = lane & 15;
    const int hi    = lane >> 4;
    const int mtile = blockIdx.x;

    const float* arow = X + (size_t)(mtile * 16 + l15) * DHID + hi * 2;
    const int col = (wave << 4) + l15;

    v8f acc = {};
    #pragma unroll
    for (int k0 = 0; k0 < DHID; k0 += 4) {
        v2f a = *(const v2f*)(arow + k0);
        const int kb = k0 + hi * 2;
        v2f b;
        b.x = W[(size_t)kb       * DHID + col];
        b.y = W[(size_t)(kb + 1) * DHID + col];
        acc = __builtin_amdgcn_wmma_f32_16x16x4_f32(
            false, a, false, b, (short)0, acc, false, false);
    }

    size_t base = (size_t)(mtile * 16 + hi * 8) * DHID + col;
    #pragma unroll
    for (int j = 0; j < 8; ++j)
        Out[base + (size_t)j * DHID] = acc[j];
}

// Self-loop folded into non-atomic init: Agg[i] = T[i] * dinv[i]^2
__global__ __launch_bounds__(256) void k_init_agg(const float* __restrict__ T,
                                                  const float* __restrict__ dinv,
                                                  float* __restrict__ Agg, int nc) {
    int i = blockIdx.x * 256 + threadIdx.x;
    if (i < nc) {
        float d = dinv[i >> 7];
        Agg[i] = T[i] * d * d;
    }
}

// One wave per edge: lane gathers float4 (coalesced 512B), 4 fp32 L2 atomics.
__global__ __launch_bounds__(256) void k_edge_agg(const int* __restrict__ src,
                                                  const int* __restrict__ dst,
                                                  const float* __restrict__ dinv,
                                                  const float* __restrict__ T,
                                                  float* __restrict__ Agg, int e) {
    int edge = (blockIdx.x * 256 + threadIdx.x) >> 5;
    if (edge >= e) return;
    int lane = threadIdx.x & 31;
    int s = src[edge], d = dst[edge];
    float nrm = dinv[s] * dinv[d];
    const float4 v = *(const float4*)(T + (size_t)s * DHID + lane * 4);
    float* o = Agg + (size_t)d * DHID + lane * 4;
    __hip_atomic_fetch_add(o + 0, v.x * nrm, __ATOMIC_RELAXED, __HIP_MEMORY_SCOPE_AGENT);
    __hip_atomic_fetch_add(o + 1, v.y * nrm, __ATOMIC_RELAXED, __HIP_MEMORY_SCOPE_AGENT);
    __hip_atomic_fetch_add(o + 2, v.z * nrm, __ATOMIC_RELAXED, __HIP_MEMORY_SCOPE_AGENT);
    __hip_atomic_fetch_add(o + 3, v.w * nrm, __ATOMIC_RELAXED, __HIP_MEMORY_SCOPE_AGENT);
}

__global__ __launch_bounds__(256) void k_bias_relu(const float* __restrict__ In,
                                                   const float* __restrict__ b,
                                                   float* __restrict__ Out, int nc) {
    int i = blockIdx.x * 256 + threadIdx.x;
    if (i < nc) Out[i] = fmaxf(In[i] + b[i & (DHID - 1)], 0.0f);
}

// 128 -> 1 readout: wave-per-node dot + wave32 shuffle reduction
__global__ __launch_bounds__(256) void k_final(const float* __restrict__ H,
                                               const float* __restrict__ Wl,
                                               const float* __restrict__ bl,
                                               float* __restrict__ out, int n) {
    int node = (blockIdx.x * 256 + threadIdx.x) >> 5;
    if (node >= n) return;
    int lane = threadIdx.x & 31;
    const float* h = H + (size_t)node * DHID;
    float s = 0.0f;
    #pragma unroll
    for (int j = 0; j < 4; ++j)
        s += h[lane + 32 * j] * Wl[lane + 32 * j];
    #pragma unroll
    for (int off = 16; off > 0; off >>= 1)
        s += __shfl_xor(s, off, 32);
    if (lane == 0) out[node] = s + bl[0];
}

extern "C" void kernel_launch(void* const* d_in, const int* in_sizes, int n_in,
                              void* d_out, int out_size, void* d_ws, size_t ws_size,
                              hipStream_t stream) {
    const float* x  = (const float*)d_in[0];
    const int*   ei = (const int*)d_in[1];   // (2,E) flat: row0=src, row1=dst
    const float* W1 = (const float*)d_in[2];
    const float* b1 = (const float*)d_in[3];
    const float* W2 = (const float*)d_in[4];
    const float* b2 = (const float*)d_in[5];
    const float* Wl = (const float*)d_in[6];
    const float* bl = (const float*)d_in[7];
    float* out = (float*)d_out;

    const int* src = ei;
    const int* dst = ei + NE;

    // workspace (floats): dinv[NN] | bufA[NN*128] | bufB[NN*128]  (~98 MB)
    float* dinv = (float*)d_ws;
    float* bufA = dinv + 100352;             // 512B-aligned element offset
    float* bufB = bufA + (size_t)NN * DHID;

    const int NC = NN * DHID;
    dim3 blk(256);
    int gN  = (NN + 255) / 256;
    int gE  = (NE + 255) / 256;
    int gNC = (NC + 255) / 256;
    int gEW = NE / 8;                        // wave per edge
    int gNW = (NN * 32 + 255) / 256;         // wave per node
    int gMM = NN / 16;                       // 6250 M-tiles (exact)

    // symmetric normalization
    k_init_deg  <<<gN,  blk, 0, stream>>>(dinv, NN);
    k_deg_atomic<<<gE,  blk, 0, stream>>>(dst, dinv, NE);
    k_dinv      <<<gN,  blk, 0, stream>>>(dinv, NN);

    // layer 1: t1 = x@W1 -> A; agg -> B; h1 = relu(agg+b1) -> A
    k_gemm_wmma <<<gMM, blk, 0, stream>>>(x, W1, bufA);
    k_init_agg  <<<gNC, blk, 0, stream>>>(bufA, dinv, bufB, NC);
    k_edge_agg  <<<gEW, blk, 0, stream>>>(src, dst, dinv, bufA, bufB, NE);
    k_bias_relu <<<gNC, blk, 0, stream>>>(bufB, b1, bufA, NC);

    // layer 2: t2 = h1@W2 -> B; agg -> A; h2 = relu(agg+b2) -> B
    k_gemm_wmma <<<gMM, blk, 0, stream>>>(bufA, W2, bufB);
    k_init_agg  <<<gNC, blk, 0, stream>>>(bufB, dinv, bufA, NC);
    k_edge_agg  <<<gEW, blk, 0, stream>>>(src, dst, dinv, bufB, bufA, NE);
    k_bias_relu <<<gNC, blk, 0, stream>>>(bufA, b2, bufB, NC);

    // readout
    k_final     <<<gNW, blk, 0, stream>>>(bufB, Wl, bl, out, NN);
}